// YOLOv1_5162550689917
// MI455X (gfx1250) — compile-verified
//
#include <hip/hip_runtime.h>

// ---------------- YOLOv1 loss constants (match reference) ----------------
#define SS 7
#define NB 2          // boxes per cell
#define NC 20         // classes
#define IMGF 448.0f
#define CELLF 64.0f   // 448/7
#define INV_CELL 0.015625f   // 1/64, exact power of two
#define INV_IMG  (1.0f/448.0f)
#define L_COORD 5.0f
#define L_NOOBJ 0.5f

static constexpr int N_SAMP = 16384;
static constexpr int N_GT   = 8;
static constexpr int N_CELL = N_SAMP * SS * SS;     // 802816
static constexpr int N_PAIR = N_SAMP * N_GT;        // 131072
static constexpr int NB1 = 1024;                    // blocks, kernel 1
static constexpr int NB2 = N_PAIR / 256;            // 512 blocks, kernel 2
static constexpr int N_PART = NB1 + NB2;            // 1536 = 24 * 64

typedef __attribute__((ext_vector_type(2))) float v2f;
typedef __attribute__((ext_vector_type(8))) float v8f;

// ---------------- Kernel 1: stream featmap, sum conf^2 (noobj base) ------
__global__ __launch_bounds__(256)
void yolo_noobj_kernel(const float* __restrict__ f, float* __restrict__ part) {
    __shared__ float sm[256];
    float acc = 0.0f;
    const int stride = gridDim.x * blockDim.x;
    for (int i = blockIdx.x * blockDim.x + threadIdx.x; i < N_CELL; i += stride) {
        const float* rowp = f + (size_t)i * 30;
        float c0 = rowp[4];
        float c1 = rowp[9];
        acc += c0 * c0 + c1 * c1;
    }
    sm[threadIdx.x] = acc;
    __syncthreads();
    for (int off = 128; off > 0; off >>= 1) {
        if (threadIdx.x < off) sm[threadIdx.x] += sm[threadIdx.x + off];
        __syncthreads();
    }
    if (threadIdx.x == 0) part[blockIdx.x] = L_NOOBJ * sm[0];
}

// ---------------- Kernel 2: per (n,g) pair terms -------------------------
__global__ __launch_bounds__(256)
void yolo_pair_kernel(const float* __restrict__ f,
                      const float* __restrict__ bb,
                      const int*   __restrict__ lab,
                      float* __restrict__ part) {
    __shared__ float sm[256];
    const int p = blockIdx.x * 256 + threadIdx.x;   // grid sized exactly: p < N_PAIR

    const float* g = bb + (size_t)p * 4;
    const float x1 = g[0], y1 = g[1], x2 = g[2], y2 = g[3];
    const float cx = 0.5f * (x1 + x2);
    const float cy = 0.5f * (y1 + y2);
    const float gw = x2 - x1;
    const float gh = y2 - y1;

    int col = (int)floorf(cx * INV_CELL);
    col = col < 0 ? 0 : (col > SS - 1 ? SS - 1 : col);
    int row = (int)floorf(cy * INV_CELL);
    row = row < 0 ? 0 : (row > SS - 1 ? SS - 1 : row);

    const int n = p / N_GT;
    const float* cell = f + ((size_t)n * (SS * SS) + (size_t)(row * SS + col)) * 30;

    const float gx0 = (float)col * CELLF;
    const float gy0 = (float)row * CELLF;
    const float a2  = fmaxf(x2 - x1, 0.0f) * fmaxf(y2 - y1, 0.0f);

    float iouv[NB], pcv[NB], pxv[NB], pyv[NB], pwv[NB], phv[NB];
#pragma unroll
    for (int b = 0; b < NB; ++b) {
        const float px = cell[5 * b + 0];
        const float py = cell[5 * b + 1];
        const float pw = cell[5 * b + 2];
        const float ph = cell[5 * b + 3];
        const float pc = cell[5 * b + 4];
        const float pcx = gx0 + px * CELLF;
        const float pcy = gy0 + py * CELLF;
        const float pwa = pw * IMGF;
        const float pha = ph * IMGF;
        const float bx1 = pcx - 0.5f * pwa, by1 = pcy - 0.5f * pha;
        const float bx2 = pcx + 0.5f * pwa, by2 = pcy + 0.5f * pha;
        const float ix1 = fmaxf(bx1, x1), iy1 = fmaxf(by1, y1);
        const float ix2 = fminf(bx2, x2), iy2 = fminf(by2, y2);
        const float inter = fmaxf(ix2 - ix1, 0.0f) * fmaxf(iy2 - iy1, 0.0f);
        const float a1 = fmaxf(bx2 - bx1, 0.0f) * fmaxf(by2 - by1, 0.0f);
        iouv[b] = inter / (a1 + a2 - inter + 1e-6f);
        pxv[b] = px; pyv[b] = py; pwv[b] = pw; phv[b] = ph; pcv[b] = pc;
    }
    // jnp.argmax: first occurrence of max -> box 1 wins only on strict >
    const int r = (iouv[1] > iouv[0]) ? 1 : 0;

    const float tx = cx * INV_CELL - (float)col;
    const float ty = cy * INV_CELL - (float)row;
    const float tw = gw * INV_IMG;
    const float th = gh * INV_IMG;

    const float dx  = pxv[r] - tx;
    const float dy  = pyv[r] - ty;
    const float dsw = sqrtf(fmaxf(pwv[r], 0.0f)) - sqrtf(tw);
    const float dsh = sqrtf(fmaxf(phv[r], 0.0f)) - sqrtf(th);
    const float coord = dx * dx + dy * dy + dsw * dsw + dsh * dsh;
    const float dconf = pcv[r] - iouv[r];

    float acc = L_COORD * coord + dconf * dconf - L_NOOBJ * pcv[r] * pcv[r];

    const int lb = lab[p];
#pragma unroll
    for (int j = 0; j < NC; ++j) {
        const float d = cell[10 + j] - (j == lb ? 1.0f : 0.0f);
        acc += d * d;
    }

    sm[threadIdx.x] = acc;
    __syncthreads();
    for (int off = 128; off > 0; off >>= 1) {
        if (threadIdx.x < off) sm[threadIdx.x] += sm[threadIdx.x + off];
        __syncthreads();
    }
    if (threadIdx.x == 0) part[NB1 + blockIdx.x] = sm[0];
}

// ---------------- Kernel 3: WMMA matrix-engine final reduction -----------
// Reduce N_PART (=24*64) f32 partials with V_WMMA_F32_16X16X4_F32:
// B = all-ones 4x16, so D[m][n] = sum_k A[m][k] + C[m][n]; every column of D
// holds the per-row sums. Total = (rows 0-7 @ col0: lane 0) + (rows 8-15 @
// col0: lane 16). Single wave32, EXEC all ones, fixed order => deterministic.
__global__ __launch_bounds__(32)
void yolo_final_kernel(const float* __restrict__ part, float* __restrict__ out) {
    __shared__ float sm[32];
    const int lane = threadIdx.x;

    v8f c = {};                 // f32 16x16 accumulator (8 VGPRs)
    v2f bm; bm[0] = 1.0f; bm[1] = 1.0f;   // ones B matrix (4x16 f32, 2 VGPRs)

    for (int i = 0; i < N_PART; i += 64) {
        v2f a;                  // A matrix 16x4 f32: 2 values per lane
        a[0] = part[i + lane];
        a[1] = part[i + 32 + lane];
        c = __builtin_amdgcn_wmma_f32_16x16x4_f32(
                /*neg_a=*/false, a, /*neg_b=*/false, bm,
                /*c_mod=*/(short)0, c, /*reuse_a=*/false, /*reuse_b=*/false);
    }

    // per-lane sum of its 8 accumulator rows (all at column == lane%16)
    float s = c[0] + c[1] + c[2] + c[3] + c[4] + c[5] + c[6] + c[7];
    sm[lane] = s;
    __syncthreads();
    if (lane == 0) out[0] = sm[0] + sm[16];   // rows 0-7 + rows 8-15, col 0
}

// ---------------- Launch --------------------------------------------------
extern "C" void kernel_launch(void* const* d_in, const int* in_sizes, int n_in,
                              void* d_out, int out_size, void* d_ws, size_t ws_size,
                              hipStream_t stream) {
    const float* featmap = (const float*)d_in[0];
    const float* bboxes  = (const float*)d_in[1];
    const int*   labels  = (const int*)d_in[2];
    float* part = (float*)d_ws;            // N_PART floats = 6 KB of scratch
    float* out  = (float*)d_out;

    yolo_noobj_kernel<<<NB1, 256, 0, stream>>>(featmap, part);
    yolo_pair_kernel <<<NB2, 256, 0, stream>>>(featmap, bboxes, labels, part);
    yolo_final_kernel<<<1, 32, 0, stream>>>(part, out);
}